// SoftSigmoid_9053791060272
// MI455X (gfx1250) — compile-verified
//
#include <hip/hip_runtime.h>

// SoftSigmoid product reduction, MI455X (gfx1250, wave32).
// Memory-bound: 256 MB streamed once -> NT b128 loads + global_prefetch.
// Transcendental-minimized: 1 exp / sigmoid element, exp+log / softplus element.
// Two-phase chunks: 8 batched NT loads (one clause), then uniform-branch compute.

typedef float v4f __attribute__((ext_vector_type(4)));

__device__ __forceinline__ float stable_softplus(float x) {
    // log1p(exp(x)) = max(x,0) + log(1 + exp(-|x|)); always finite.
    float e = __expf(-__builtin_fabsf(x));
    return fmaxf(x, 0.0f) + __logf(1.0f + e);
}

__global__ __launch_bounds__(256) void soft_sigmoid_prod64(
    const float* __restrict__ x, const int* __restrict__ cidx, int n_idx,
    float* __restrict__ out, int cols)
{
    constexpr int ROWS = 64;
    constexpr int CHUNK = 8;
    long long c = ((long long)blockIdx.x * blockDim.x + threadIdx.x) * 4;
    if (c >= cols) return;

    // Build convex-row bitmask; force into SGPRs (uniform across the wave)
    // so the per-row branch lowers to a scalar branch, not exec masking.
    unsigned long long mask = 0ull;
    for (int j = 0; j < n_idx; ++j) mask |= 1ull << (cidx[j] & 63);
    unsigned mlo = __builtin_amdgcn_readfirstlane((unsigned)mask);
    unsigned mhi = __builtin_amdgcn_readfirstlane((unsigned)(mask >> 32));
    mask = ((unsigned long long)mhi << 32) | mlo;

    const float* base = x + c;
    const size_t stride = (size_t)cols;
    v4f pn = {1.f, 1.f, 1.f, 1.f};   // product of softplus values
    v4f pd = {1.f, 1.f, 1.f, 1.f};   // product of (1 + e^{-x}) for sigmoid rows

    for (int r0 = 0; r0 < ROWS; r0 += CHUNK) {
        // Phase 1: batch all 8 NT loads -> single load clause, one wait.
        v4f v[CHUNK];
#pragma unroll
        for (int k = 0; k < CHUNK; ++k)
            v[k] = __builtin_nontemporal_load(
                (const v4f*)(base + (size_t)(r0 + k) * stride)); // global_load_b128 NT

        // Phase 2: prefetch two chunks ahead (64 MB lead in the stream).
#pragma unroll
        for (int k = 0; k < CHUNK; ++k) {
            int pr = r0 + k + 2 * CHUNK;
            if (pr > ROWS - 1) pr = ROWS - 1;
            __builtin_prefetch(base + (size_t)pr * stride, 0, 1); // global_prefetch_b8
        }

        // Phase 3: branchy compute, uniform per row (scalar branch, no EXEC churn).
#pragma unroll
        for (int k = 0; k < CHUNK; ++k) {
            if ((mask >> (r0 + k)) & 1ull) {
                pn.x *= stable_softplus(v[k].x);
                pn.y *= stable_softplus(v[k].y);
                pn.z *= stable_softplus(v[k].z);
                pn.w *= stable_softplus(v[k].w);
            } else {
                pd.x *= 1.0f + __expf(-v[k].x);
                pd.y *= 1.0f + __expf(-v[k].y);
                pd.z *= 1.0f + __expf(-v[k].z);
                pd.w *= 1.0f + __expf(-v[k].w);
            }
        }
    }

    v4f res = pn / pd;                                  // one precise divide per column
    __builtin_nontemporal_store(res, (v4f*)(out + c));
}

// Fallback for non-64-row / non-multiple-of-4-col shapes (not hit by the
// reference harness, but keeps kernel_launch deterministic & general).
__global__ __launch_bounds__(256) void soft_sigmoid_prod_gen(
    const float* __restrict__ x, const int* __restrict__ cidx, int n_idx,
    float* __restrict__ out, int rows, int cols)
{
    long long c = (long long)blockIdx.x * blockDim.x + threadIdx.x;
    if (c >= cols) return;
    float pn = 1.0f, pd = 1.0f;
    for (int r = 0; r < rows; ++r) {
        bool convex = false;
        for (int j = 0; j < n_idx; ++j) convex |= (cidx[j] == r);
        float v = x[(size_t)r * (size_t)cols + c];
        if (convex) pn *= stable_softplus(v);
        else        pd *= 1.0f + __expf(-v);
    }
    out[c] = pn / pd;
}

extern "C" void kernel_launch(void* const* d_in, const int* in_sizes, int n_in,
                              void* d_out, int out_size, void* d_ws, size_t ws_size,
                              hipStream_t stream) {
    const float* x    = (const float*)d_in[0];
    const int*   cidx = (const int*)d_in[1];
    const int n_idx   = in_sizes[1];
    const int cols    = out_size;                 // output is one value per column
    const int rows    = (cols > 0) ? (in_sizes[0] / cols) : 0;
    float* out = (float*)d_out;

    if (rows == 64 && cols % 4 == 0 && cols > 0) {
        int threads = cols / 4;
        dim3 block(256);
        dim3 grid((threads + 255) / 256);
        soft_sigmoid_prod64<<<grid, block, 0, stream>>>(x, cidx, n_idx, out, cols);
    } else if (cols > 0) {
        dim3 block(256);
        dim3 grid((cols + 255) / 256);
        soft_sigmoid_prod_gen<<<grid, block, 0, stream>>>(x, cidx, n_idx, out, rows, cols);
    }
}